// Model_59090160058943
// MI455X (gfx1250) — compile-verified
//
#include <hip/hip_runtime.h>
#include <math.h>
#include <limits.h>

namespace {

constexpr int kN    = 200000;   // nodes
constexpr int kE    = 3200000;  // edges
constexpr int kFin  = 128;
constexpr int kH    = 64;
constexpr int kOut  = 2;
constexpr int kB    = 128;      // graphs
constexpr int kK    = 64;       // sort-pool k
constexpr float kAlpha = 0.1f;
constexpr int kHops = 10;

typedef float v2f __attribute__((ext_vector_type(2)));
typedef float v8f __attribute__((ext_vector_type(8)));

// ---------------- degree / GCN norm ----------------

__global__ void init_deg_kernel(float* __restrict__ deg) {
  int i = blockIdx.x * blockDim.x + threadIdx.x;
  if (i < kN) deg[i] = 1.0f;  // self loop
}

__global__ void accum_deg_kernel(const int* __restrict__ dst, float* __restrict__ deg) {
  int e = blockIdx.x * blockDim.x + threadIdx.x;
  if (e < kE) atomicAdd(&deg[dst[e]], 1.0f);
}

__global__ void dinv_kernel(const float* __restrict__ deg, float* __restrict__ dinv) {
  int i = blockIdx.x * blockDim.x + threadIdx.x;
  if (i < kN) dinv[i] = rsqrtf(deg[i]);
}

__global__ void norm_kernel(const int* __restrict__ src, const int* __restrict__ dst,
                            const float* __restrict__ dinv, float* __restrict__ norm) {
  int e = blockIdx.x * blockDim.x + threadIdx.x;
  if (e < kE) norm[e] = dinv[src[e]] * dinv[dst[e]];
}

// ---------------- h0 = X @ W1 + b1 via fp32 WMMA ----------------
// One wave computes one 16x16 output tile; K=128 in 32 steps of 16x16x4.
// Block = 128 threads = 4 waves = 4 column tiles (64 = 4*16 columns).

__global__ void gemm_wmma_kernel(const float* __restrict__ X, const float* __restrict__ W,
                                 const float* __restrict__ bias, float* __restrict__ out) {
  const int rowTile = blockIdx.x;           // N/16 tiles
  const int wave    = threadIdx.x >> 5;     // 0..3 -> column tile
  const int lane    = threadIdx.x & 31;
  const int m       = lane & 15;
  const int khalf   = (lane >> 4) << 1;     // 0 or 2 (A/B frag K pair)
  const float* xrow = X + (size_t)(rowTile * 16 + m) * kFin;
  const int nb      = wave * 16 + m;        // B frag / output column

  v8f c = {};
  for (int k = 0; k < kFin; k += 4) {
    v2f a, b;
    a[0] = xrow[k + khalf];
    a[1] = xrow[k + khalf + 1];
    b[0] = W[(size_t)(k + khalf) * kH + nb];
    b[1] = W[(size_t)(k + khalf + 1) * kH + nb];
    c = __builtin_amdgcn_wmma_f32_16x16x4_f32(false, a, false, b, (short)0, c,
                                              false, false);
  }

  const float bv   = bias[nb];
  const int  rbase = (lane >> 4) << 3;      // VGPR r holds M=r (lanes 0-15) / M=r+8
#pragma unroll
  for (int r = 0; r < 8; ++r) {
    int row = rowTile * 16 + rbase + r;
    out[(size_t)row * kH + nb] = c[r] + bv;
  }
}

// ---------------- APPNP, 64 features ----------------
// z_{t+1} = 0.9 * (A_hat z_t) + 0.1 * h ; self-loop term folded into init.

__global__ void appnp_init64_kernel(const float* __restrict__ h, const float* __restrict__ z,
                                    const float* __restrict__ dinv, float* __restrict__ zn) {
  int t = blockIdx.x * blockDim.x + threadIdx.x;  // kN*16 threads, 4 floats each
  if (t >= kN * 16) return;
  int node = t >> 4;
  int off  = (t & 15) << 2;
  float d2 = dinv[node];
  d2 = 0.9f * d2 * d2;                             // (1-alpha) * selfloop norm
  const float4 hv = *(const float4*)(h + (size_t)node * kH + off);
  const float4 zv = *(const float4*)(z + (size_t)node * kH + off);
  float4 o;
  o.x = kAlpha * hv.x + d2 * zv.x;
  o.y = kAlpha * hv.y + d2 * zv.y;
  o.z = kAlpha * hv.z + d2 * zv.z;
  o.w = kAlpha * hv.w + d2 * zv.w;
  *(float4*)(zn + (size_t)node * kH + off) = o;
}

__global__ void appnp_edge64_kernel(const int* __restrict__ src, const int* __restrict__ dst,
                                    const float* __restrict__ norm,
                                    const float* __restrict__ z, float* __restrict__ zn) {
  int t = blockIdx.x * blockDim.x + threadIdx.x;  // kE*16 threads, 4 floats each
  if (t >= kE * 16) return;
  int e   = t >> 4;
  int off = (t & 15) << 2;
  float w = 0.9f * norm[e];
  const float4 zv = *(const float4*)(z + (size_t)src[e] * kH + off);
  float* p = zn + (size_t)dst[e] * kH + off;
  atomicAdd(p + 0, w * zv.x);
  atomicAdd(p + 1, w * zv.y);
  atomicAdd(p + 2, w * zv.z);
  atomicAdd(p + 3, w * zv.w);
}

// ---------------- h1 = z @ W2 + b2 (64 -> 2) ----------------

__global__ void lin2_kernel(const float* __restrict__ z, const float* __restrict__ W2,
                            const float* __restrict__ b2, float* __restrict__ h1) {
  int i = blockIdx.x * blockDim.x + threadIdx.x;
  if (i >= kN) return;
  float a0 = b2[0], a1 = b2[1];
  const float* zr = z + (size_t)i * kH;
#pragma unroll 8
  for (int k = 0; k < kH; ++k) {
    float v = zr[k];
    a0 = fmaf(v, W2[2 * k + 0], a0);
    a1 = fmaf(v, W2[2 * k + 1], a1);
  }
  h1[2 * i + 0] = a0;
  h1[2 * i + 1] = a1;
}

// ---------------- APPNP, 2 features ----------------

__global__ void appnp_init2_kernel(const float* __restrict__ h, const float* __restrict__ z,
                                   const float* __restrict__ dinv, float* __restrict__ zn) {
  int i = blockIdx.x * blockDim.x + threadIdx.x;
  if (i >= kN) return;
  float d2 = dinv[i];
  d2 = 0.9f * d2 * d2;
  zn[2 * i + 0] = kAlpha * h[2 * i + 0] + d2 * z[2 * i + 0];
  zn[2 * i + 1] = kAlpha * h[2 * i + 1] + d2 * z[2 * i + 1];
}

__global__ void appnp_edge2_kernel(const int* __restrict__ src, const int* __restrict__ dst,
                                   const float* __restrict__ norm,
                                   const float* __restrict__ z, float* __restrict__ zn) {
  int e = blockIdx.x * blockDim.x + threadIdx.x;
  if (e >= kE) return;
  float w = 0.9f * norm[e];
  int s = src[e], d = dst[e];
  atomicAdd(&zn[2 * d + 0], w * z[2 * s + 0]);
  atomicAdd(&zn[2 * d + 1], w * z[2 * s + 1]);
}

// ---------------- batch counts / starts ----------------

__global__ void count_kernel(const int* __restrict__ batch, int* __restrict__ counts) {
  int i = blockIdx.x * blockDim.x + threadIdx.x;
  if (i < kN) atomicAdd(&counts[batch[i]], 1);
}

__global__ void starts_kernel(const int* __restrict__ counts, int* __restrict__ starts) {
  if (blockIdx.x == 0 && threadIdx.x == 0) {
    int acc = 0;
    for (int g = 0; g < kB; ++g) { starts[g] = acc; acc += counts[g]; }
  }
}

// ---------------- global sort pool: top-64 by (key desc, idx asc) ----------------
// batch is sorted, so graph g owns nodes [starts[g], starts[g]+counts[g]).
// Iterative selection reproduces stable lexsort((-key, batch)) tie-breaking.

__global__ void sortpool_kernel(const float* __restrict__ h, const int* __restrict__ starts,
                                const int* __restrict__ counts, float* __restrict__ pooled) {
  __shared__ float sk[256];
  __shared__ int   si[256];
  const int g   = blockIdx.x;
  const int tid = threadIdx.x;
  const int s   = starts[g];
  const int cnt = counts[g];

  float prevK = __builtin_inff();
  int   prevI = -1;

  for (int r = 0; r < kK; ++r) {
    float bestK = -__builtin_inff();
    int   bestI = INT_MAX;
    for (int j = s + tid; j < s + cnt; j += blockDim.x) {
      float key = h[2 * j + 1];  // last channel is the sort key
      bool after = (key < prevK) || (key == prevK && j > prevI);
      if (after && (key > bestK || (key == bestK && j < bestI))) {
        bestK = key; bestI = j;
      }
    }
    sk[tid] = bestK; si[tid] = bestI;
    __syncthreads();
    for (int offs = 128; offs > 0; offs >>= 1) {
      if (tid < offs) {
        float ok = sk[tid + offs]; int oi = si[tid + offs];
        if (ok > sk[tid] || (ok == sk[tid] && oi < si[tid])) { sk[tid] = ok; si[tid] = oi; }
      }
      __syncthreads();
    }
    float wk = sk[0]; int wi = si[0];
    __syncthreads();
    if (wi == INT_MAX) break;   // fewer than k nodes; rest stays zero-padded
    prevK = wk; prevI = wi;
    if (tid == 0) {
      pooled[g * (kK * kOut) + 2 * r + 0] = h[2 * wi];
      pooled[g * (kK * kOut) + 2 * r + 1] = wk;
    }
  }
}

// ---------------- final linear [B,128] @ [128,1] + b ----------------

__global__ void final_kernel(const float* __restrict__ pooled, const float* __restrict__ Wlin,
                             const float* __restrict__ blin, float* __restrict__ out) {
  int g = threadIdx.x;
  if (g < kB) {
    float acc = blin[0];
    const float* pr = pooled + (size_t)g * (kK * kOut);
#pragma unroll 8
    for (int j = 0; j < kK * kOut; ++j) acc = fmaf(pr[j], Wlin[j], acc);
    out[g] = acc;
  }
}

}  // namespace

extern "C" void kernel_launch(void* const* d_in, const int* in_sizes, int n_in,
                              void* d_out, int out_size, void* d_ws, size_t ws_size,
                              hipStream_t stream) {
  const float* x     = (const float*)d_in[0];
  const int*   edge  = (const int*)d_in[1];   // [2, E]
  const int*   batch = (const int*)d_in[2];
  const float* W1    = (const float*)d_in[3];
  const float* b1    = (const float*)d_in[4];
  const float* W2    = (const float*)d_in[5];
  const float* b2    = (const float*)d_in[6];
  const float* Wlin  = (const float*)d_in[7];
  const float* blin  = (const float*)d_in[8];
  float* out = (float*)d_out;

  const int* src = edge;
  const int* dst = edge + kE;

  // workspace layout
  float* p      = (float*)d_ws;
  float* deg    = p; p += kN;
  float* dinv   = p; p += kN;
  float* norm   = p; p += kE;
  float* h0     = p; p += (size_t)kN * kH;
  float* za     = p; p += (size_t)kN * kH;
  float* zb     = p; p += (size_t)kN * kH;
  float* h1     = p; p += (size_t)kN * kOut;
  float* z2a    = p; p += (size_t)kN * kOut;
  float* z2b    = p; p += (size_t)kN * kOut;
  float* pooled = p; p += (size_t)kB * kK * kOut;
  int*   counts = (int*)p;
  int*   starts = counts + kB;

  const dim3 b256(256);

  // GCN normalization
  init_deg_kernel <<<(kN + 255) / 256, b256, 0, stream>>>(deg);
  accum_deg_kernel<<<(kE + 255) / 256, b256, 0, stream>>>(dst, deg);
  dinv_kernel     <<<(kN + 255) / 256, b256, 0, stream>>>(deg, dinv);
  norm_kernel     <<<(kE + 255) / 256, b256, 0, stream>>>(src, dst, dinv, norm);

  // h0 = x @ W1 + b1  (fp32 WMMA; N = 12500 * 16 exactly)
  gemm_wmma_kernel<<<kN / 16, 128, 0, stream>>>(x, W1, b1, h0);

  // APPNP #1 (64 features), ping-pong z buffers, z starts at h0
  const float* z  = h0;
  float*       zn = za;
  for (int hop = 0; hop < kHops; ++hop) {
    appnp_init64_kernel<<<(kN * 16 + 255) / 256, b256, 0, stream>>>(h0, z, dinv, zn);
    appnp_edge64_kernel<<<(kE * 16 + 255) / 256, b256, 0, stream>>>(src, dst, norm, z, zn);
    z  = zn;
    zn = (zn == za) ? zb : za;
  }

  // h1 = z @ W2 + b2
  lin2_kernel<<<(kN + 255) / 256, b256, 0, stream>>>(z, W2, b2, h1);

  // APPNP #2 (2 features)
  const float* z2  = h1;
  float*       z2n = z2a;
  for (int hop = 0; hop < kHops; ++hop) {
    appnp_init2_kernel<<<(kN + 255) / 256, b256, 0, stream>>>(h1, z2, dinv, z2n);
    appnp_edge2_kernel<<<(kE + 255) / 256, b256, 0, stream>>>(src, dst, norm, z2, z2n);
    z2  = z2n;
    z2n = (z2n == z2a) ? z2b : z2a;
  }

  // per-graph counts / starts (batch is sorted)
  hipMemsetAsync(counts, 0, kB * sizeof(int), stream);
  count_kernel <<<(kN + 255) / 256, b256, 0, stream>>>(batch, counts);
  starts_kernel<<<1, 32, 0, stream>>>(counts, starts);

  // sort pool (zero-pad first)
  hipMemsetAsync(pooled, 0, (size_t)kB * kK * kOut * sizeof(float), stream);
  sortpool_kernel<<<kB, 256, 0, stream>>>(z2, starts, counts, pooled);

  // final linear
  final_kernel<<<1, 128, 0, stream>>>(pooled, Wlin, blin, out);
}